// HetGraphLayer_31387620999790
// MI455X (gfx1250) — compile-verified
//
#include <hip/hip_runtime.h>

// HetGraphLayer: out = mean_r( x@Ws_r + mean-neigh_r@Wn_r + b_r )
// Fused as: out = x@(mean_r Ws_r) + sum_r (neigh_r * 1/(R*cnt)) @ Wn_r + mean_r b_r
// fp32 WMMA (v_wmma_f32_16x16x4_f32); per-relation scatter via L2 atomics;
// W staged to LDS via gfx1250 async global->LDS loads (ASYNCcnt).

typedef __attribute__((ext_vector_type(2))) float v2f;
typedef __attribute__((ext_vector_type(8))) float v8f;

#define D    128
#define TILE 16

// XOR swizzle so rows k and k+2 (read by the two half-waves of one wave in the
// same LDS load) land in disjoint 16-bank windows of the 64 LDS banks.
// Only permutes 16-col windows -> any aligned 16B float4 stays contiguous.
__device__ __forceinline__ int swz_idx(int k, int j) {
    return k * D + (j ^ (((k >> 1) & 3) << 4));
}

// out[i] = invR * sum_r b[r*D + (i mod D)]
__global__ __launch_bounds__(256) void init_out_kernel(const float* __restrict__ b,
                                                       float* __restrict__ out,
                                                       long total, int R, float invR) {
    long i = (long)blockIdx.x * blockDim.x + threadIdx.x;
    if (i >= total) return;
    int d = (int)(i & (D - 1));
    float s = 0.f;
    for (int r = 0; r < R; ++r) s += b[(size_t)r * D + d];
    out[i] = s * invR;
}

// Wsum = invR * sum_r Wself[r]
__global__ __launch_bounds__(256) void wsum_kernel(const float* __restrict__ Wself,
                                                   float* __restrict__ Wsum,
                                                   int R, float invR) {
    int i = blockIdx.x * blockDim.x + threadIdx.x;
    if (i >= D * D) return;
    float s = 0.f;
    for (int r = 0; r < R; ++r) s += Wself[(size_t)r * D * D + i];
    Wsum[i] = s * invR;
}

__global__ __launch_bounds__(256) void zero_kernel(float* __restrict__ p, long n) {
    long i = (long)blockIdx.x * blockDim.x + threadIdx.x;
    if (i < n) p[i] = 0.f;
}

// One wave (32 lanes) per edge: gather a 128-float row of x coalesced
// (float4 per lane) and atomic-add into neigh[dst]; lane 0 bumps cnt[dst].
// neigh (51.2 MB) + x (51.2 MB) are L2-resident on MI455X (192 MB L2), so
// these atomics resolve in L2 atomic units, not HBM RMW.
__global__ __launch_bounds__(256) void scatter_kernel(const float* __restrict__ x,
                                                      const int* __restrict__ src,
                                                      const int* __restrict__ dst,
                                                      float* __restrict__ neigh,
                                                      float* __restrict__ cnt,
                                                      int nE) {
    int wave = blockIdx.x * (blockDim.x >> 5) + (threadIdx.x >> 5);
    int lane = threadIdx.x & 31;
    if (wave >= nE) return;
    int s = src[wave];
    int d = dst[wave];
    const float4* xr = (const float4*)(x + (size_t)s * D);
    float4 v = xr[lane];
    float* nb = neigh + (size_t)d * D + lane * 4;
    atomicAdd(nb + 0, v.x);
    atomicAdd(nb + 1, v.y);
    atomicAdd(nb + 2, v.z);
    atomicAdd(nb + 3, v.w);
    if (lane == 0) atomicAdd(cnt + d, 1.0f);
}

// out[M,128] += diag(scale) * A[M,128] @ W[128,128]
//   scale = 1                          if cnt == nullptr   (x @ Wsum path)
//   scale = invR / max(cnt[row], 1)    otherwise           (neigh mean path)
// 256 threads = 8 waves; W async-staged into LDS (64 KB, swizzled); each wave
// owns one 16-row stripe x all 8 column tiles: 32 K-steps x 8 tiles of
// v_wmma_f32_16x16x4_f32, accumulators in 64 VGPRs.
__global__ __launch_bounds__(256) void gemm_acc_kernel(const float* __restrict__ A,
                                                       const float* __restrict__ W,
                                                       const float* __restrict__ cnt,
                                                       float* __restrict__ out,
                                                       int ntiles, float invR) {
    __shared__ float lB[D * D];  // exactly 64 KB
    int tid = threadIdx.x;

    // Async global->LDS staging of W (gfx1250 path, tracked by ASYNCcnt).
    // Generic AS pointers to LDS carry the LDS byte offset in addr[31:0]
    // (ISA 10.2 aperture mapping), so truncation gives the raw LDS address
    // each lane supplies as the async load's per-lane destination.
    {
        unsigned lds_base = (unsigned)(uintptr_t)(&lB[0]);
        #pragma unroll
        for (int i = 0; i < 16; ++i) {        // 4096 float4s / 256 threads
            int f  = tid + i * 256;           // float4 index
            int k  = f >> 5;                  // row of W
            int c4 = (f & 31) << 2;           // first column of the float4
            int m  = ((k >> 1) & 3) << 4;     // swizzle mask (multiple of 16)
            unsigned lds_off = lds_base + (unsigned)((k * D + (c4 ^ m)) * 4);
            const float* gsrc = W + (size_t)f * 4;
            asm volatile("global_load_async_to_lds_b128 %0, %1, off"
                         :: "v"(lds_off), "v"(gsrc)
                         : "memory");
        }
        asm volatile("s_wait_asynccnt 0x0" ::: "memory");
    }
    __syncthreads();

    int wave = tid >> 5;
    int lane = tid & 31;
    int half = lane >> 4;     // which half-wave (K split for A/B fragments)
    int l15  = lane & 15;     // M index (A) / N index (B)
    int row_tile = blockIdx.x * 8 + wave;
    if (row_tile >= ntiles) return;   // wave-uniform: EXEC stays all-1s inside

    int m0   = row_tile * TILE;
    int mrow = m0 + l15;
    float scale = 1.0f;
    if (cnt) scale = invR / fmaxf(cnt[mrow], 1.0f);
    // A 16x4 fragment: lane (l15,half) supplies A[mrow][4*k4 + 2*half + {0,1}]
    // -> exactly one float2 per K-step: index 2*k4 + half.
    const float2* arow2 = (const float2*)(A + (size_t)mrow * D);

    v8f c[8] = {};
    for (int k4 = 0; k4 < 32; ++k4) {
        float2 av = arow2[2 * k4 + half];
        v2f a;
        a.x = av.x * scale;
        a.y = av.y * scale;
        int r0 = 4 * k4 + 2 * half;  // B rows for this half-wave: r0, r0+1
        #pragma unroll
        for (int ct = 0; ct < 8; ++ct) {
            int col = ct * TILE + l15;
            v2f bb;
            bb.x = lB[swz_idx(r0,     col)];
            bb.y = lB[swz_idx(r0 + 1, col)];
            c[ct] = __builtin_amdgcn_wmma_f32_16x16x4_f32(
                        false, a, false, bb, (short)0, c[ct], false, false);
        }
    }

    // C/D layout: VGPR j -> row m0 + j + 8*half, col = ct*16 + l15.
    int orow = m0 + 8 * half;
    #pragma unroll
    for (int ct = 0; ct < 8; ++ct) {
        #pragma unroll
        for (int j = 0; j < 8; ++j) {
            size_t o = (size_t)(orow + j) * D + ct * TILE + l15;
            out[o] += c[ct][j];
        }
    }
}

extern "C" void kernel_launch(void* const* d_in, const int* in_sizes, int n_in,
                              void* d_out, int out_size, void* d_ws, size_t ws_size,
                              hipStream_t stream) {
    const float* x      = (const float*)d_in[0];
    const int*   src    = (const int*)d_in[1];
    const int*   dst    = (const int*)d_in[2];
    const float* Wself  = (const float*)d_in[3];
    const float* Wneigh = (const float*)d_in[4];
    const float* b      = (const float*)d_in[5];
    float* out = (float*)d_out;

    int N = in_sizes[0] / D;          // 100000
    int R = in_sizes[5] / D;          // 4
    int E = in_sizes[1] / R;          // 500000
    float invR = 1.0f / (float)R;

    // Workspace: neigh [N,128] | cnt [N] | Wsum [128,128]  (~51.7 MB)
    float* neigh = (float*)d_ws;
    float* cnt   = neigh + (size_t)N * D;
    float* Wsum  = cnt + N;

    long totalOut  = (long)N * D;
    int  ntiles    = N / TILE;                 // 6250 (N divisible by 16)
    int  gemm_blks = (ntiles + 7) / 8;

    wsum_kernel<<<(D * D + 255) / 256, 256, 0, stream>>>(Wself, Wsum, R, invR);
    init_out_kernel<<<(int)((totalOut + 255) / 256), 256, 0, stream>>>(
        b, out, totalOut, R, invR);
    gemm_acc_kernel<<<gemm_blks, 256, 0, stream>>>(x, Wsum, nullptr, out,
                                                   ntiles, invR);

    long zn = (long)N * D + N;   // neigh + cnt are contiguous
    for (int r = 0; r < R; ++r) {
        zero_kernel<<<(int)((zn + 255) / 256), 256, 0, stream>>>(neigh, zn);
        scatter_kernel<<<(E + 7) / 8, 256, 0, stream>>>(
            x, src + (size_t)r * E, dst + (size_t)r * E, neigh, cnt, E);
        gemm_acc_kernel<<<gemm_blks, 256, 0, stream>>>(
            neigh, Wneigh + (size_t)r * D * D, cnt, out, ntiles, invR);
    }
}